// QASALayer_81939386073592
// MI455X (gfx1250) — compile-verified
//
#include <hip/hip_runtime.h>

// ---------------------------------------------------------------------------
// QASA layer for MI455X (gfx1250, wave32):
//   k0: pad W_in [8,1024] -> [16,1024] (zero rows 8..15) so WMMA B-frag loads
//       are unpredicated (EXEC stays all-1s through v_wmma).
//   k1: x_proj = x @ W_in^T + b_in via V_WMMA_F32_16X16X4_F32, 1 wave / 16 rows.
//   k2: fused 8-qubit statevector sim (1 sample / wave, 8 complex amps per
//       lane, shfl_xor butterflies for lane-bit wires) + expval reduction +
//       output GEMM (K=8) + bias, coalesced float4 stores.
// ---------------------------------------------------------------------------

typedef float v2f __attribute__((ext_vector_type(2)));
typedef float v8f __attribute__((ext_vector_type(8)));

#define NQ      8
#define NLAYERS 4
#define DIN     1024
#define DOUT    1024

// ------------------------------- kernel 0 ----------------------------------
__global__ __launch_bounds__(256) void qsa_pad_win(const float* __restrict__ Win,
                                                   float* __restrict__ wpad) {
  int idx = blockIdx.x * 256 + threadIdx.x;   // 0 .. 16*1024-1
  int n = idx >> 10;
  int k = idx & 1023;
  wpad[idx] = (n < NQ) ? Win[n * DIN + k] : 0.0f;
}

// ------------------------------- kernel 1 ----------------------------------
// One wave per 16-row tile of x. K accumulated 4 at a time with f32 WMMA.
__global__ __launch_bounds__(32) void qsa_gemm1_wmma(
    const float* __restrict__ x,      // [B, 1024]
    const float* __restrict__ wpad,   // [16, 1024]  (B matrix, rows = N)
    const float* __restrict__ bin,    // [8]
    float* __restrict__ xproj) {      // [B, 8]
  const int lane = threadIdx.x;       // 0..31
  const int tile = blockIdx.x;
  const int hk   = lane >> 4;         // half-wave: K offset 0 or 2
  const int mn   = lane & 15;         // A: M row, B: N col

  const float* xrow = x    + (size_t)(tile * 16 + mn) * DIN;
  const float* wrow = wpad + (size_t)mn * DIN;

  v8f acc0 = {};
  v8f acc1 = {};
  for (int k = 0; k < DIN; k += 8) {
    const int ka = k + hk * 2;
    v2f a0 = *(const v2f*)(xrow + ka);
    v2f b0 = *(const v2f*)(wrow + ka);
    v2f a1 = *(const v2f*)(xrow + ka + 4);
    v2f b1 = *(const v2f*)(wrow + ka + 4);
    acc0 = __builtin_amdgcn_wmma_f32_16x16x4_f32(false, a0, false, b0,
                                                 (short)0, acc0, false, false);
    acc1 = __builtin_amdgcn_wmma_f32_16x16x4_f32(false, a1, false, b1,
                                                 (short)0, acc1, false, false);
  }
  v8f acc = acc0 + acc1;

  // C layout: vgpr v -> (M = v or v+8 per half-wave, N = lane&15)
  if (mn < NQ) {
    const float bb = bin[mn];
#pragma unroll
    for (int v = 0; v < 8; ++v) {
      const int m = hk ? (v + 8) : v;
      xproj[(size_t)(tile * 16 + m) * NQ + mn] = acc[v] + bb;
    }
  }
}

// ------------------------------- kernel 2 ----------------------------------
// Index convention: basis index idx = lane*8 + r  (idx bit b: b>=3 -> lane bit
// b-3, b<3 -> register bit). Wire w corresponds to bit (7-w).

__device__ __forceinline__ float sx32(float v, int m) { return __shfl_xor(v, m, 32); }

template <int BBIT>
__device__ __forceinline__ void rx_gate(float (&ar)[8], float (&ai)[8],
                                        float c, float s, int lane) {
  (void)lane;
  if constexpr (BBIT >= 3) {
    const int mask = 1 << (BBIT - 3);
#pragma unroll
    for (int r = 0; r < 8; ++r) {   // RX matrix is symmetric: a' = c*a - i*s*p
      float pr = sx32(ar[r], mask);
      float pi = sx32(ai[r], mask);
      float nr = c * ar[r] + s * pi;
      float ni = c * ai[r] - s * pr;
      ar[r] = nr; ai[r] = ni;
    }
  } else {
    const int m = 1 << BBIT;
#pragma unroll
    for (int r = 0; r < 8; ++r) {
      if ((r & m) == 0) {
        const int r2 = r | m;
        float x0 = ar[r], y0 = ai[r], x1 = ar[r2], y1 = ai[r2];
        ar[r]  = c * x0 + s * y1;  ai[r]  = c * y0 - s * x1;
        ar[r2] = c * x1 + s * y0;  ai[r2] = c * y1 - s * x0;
      }
    }
  }
}

template <int BBIT>
__device__ __forceinline__ void rz_gate(float (&ar)[8], float (&ai)[8],
                                        float c, float s, int lane) {
  if constexpr (BBIT >= 3) {
    const float t = ((lane >> (BBIT - 3)) & 1) ? s : -s;
#pragma unroll
    for (int r = 0; r < 8; ++r) {
      float x = ar[r], y = ai[r];
      ar[r] = c * x - t * y;  ai[r] = c * y + t * x;
    }
  } else {
#pragma unroll
    for (int r = 0; r < 8; ++r) {
      const float t = (r & (1 << BBIT)) ? s : -s;   // compile-time sign
      float x = ar[r], y = ai[r];
      ar[r] = c * x - t * y;  ai[r] = c * y + t * x;
    }
  }
}

template <int BC, int BT>  // control bit, target bit (BC = BT+1)
__device__ __forceinline__ void cnot_gate(float (&ar)[8], float (&ai)[8], int lane) {
  if constexpr (BT >= 3) {                 // both lane bits: masked shfl swap
    const int mask = 1 << (BT - 3);
    const bool ctrl = ((lane >> (BC - 3)) & 1) != 0;
#pragma unroll
    for (int r = 0; r < 8; ++r) {
      float pr = sx32(ar[r], mask);
      float pi = sx32(ai[r], mask);
      ar[r] = ctrl ? pr : ar[r];
      ai[r] = ctrl ? pi : ai[r];
    }
  } else if constexpr (BC >= 3) {          // BC==3 (lane bit 0), BT==2 (reg)
    const bool ctrl = (lane & 1) != 0;
#pragma unroll
    for (int r = 0; r < 4; ++r) {
      float a = ar[r], b = ar[r + 4];
      ar[r]     = ctrl ? b : a;
      ar[r + 4] = ctrl ? a : b;
      float cc = ai[r], d = ai[r + 4];
      ai[r]     = ctrl ? d : cc;
      ai[r + 4] = ctrl ? cc : d;
    }
  } else {                                 // both register bits: free renames
#pragma unroll
    for (int r = 0; r < 8; ++r) {
      if ((r & (1 << BC)) && !(r & (1 << BT))) {
        const int r2 = r | (1 << BT);
        float t = ar[r]; ar[r] = ar[r2]; ar[r2] = t;
        t = ai[r]; ai[r] = ai[r2]; ai[r2] = t;
      }
    }
  }
}

__global__ __launch_bounds__(256) void qsa_qsim_out(
    const float* __restrict__ xproj,  // [B, 8]
    const float* __restrict__ qw,     // [4, 2, 8]
    const float* __restrict__ Wout,   // [1024, 8]
    const float* __restrict__ bout,   // [1024]
    float* __restrict__ out) {        // [B, 1024]
  const int lane = threadIdx.x & 31;
  const int row  = blockIdx.x * 8 + (threadIdx.x >> 5);

  float th[NQ];
#pragma unroll
  for (int i = 0; i < NQ; ++i) th[i] = xproj[(size_t)row * NQ + i];

  float ar[8], ai[8];
#pragma unroll
  for (int r = 0; r < 8; ++r) { ar[r] = 0.0f; ai[r] = 0.0f; }
  ar[0] = (lane == 0) ? 1.0f : 0.0f;   // |0...0>

#define QSA_INIT(W) do { float s1, c1;                               \
    __sincosf(0.5f * th[(W)], &s1, &c1);                             \
    rx_gate<7 - (W)>(ar, ai, c1, s1, lane);                          \
    rz_gate<7 - (W)>(ar, ai, c1, s1, lane); } while (0)

  QSA_INIT(0); QSA_INIT(1); QSA_INIT(2); QSA_INIT(3);
  QSA_INIT(4); QSA_INIT(5); QSA_INIT(6); QSA_INIT(7);

#define QSA_RXRZ(W, L) do { float s1, c1, s2, c2;                    \
    __sincosf(0.5f * qw[(L) * 16 + (W)], &s1, &c1);                  \
    __sincosf(0.5f * qw[(L) * 16 + 8 + (W)], &s2, &c2);              \
    rx_gate<7 - (W)>(ar, ai, c1, s1, lane);                          \
    rz_gate<7 - (W)>(ar, ai, c2, s2, lane); } while (0)

#define QSA_LAYER(L) do {                                            \
    QSA_RXRZ(0, L); QSA_RXRZ(1, L); QSA_RXRZ(2, L); QSA_RXRZ(3, L);  \
    QSA_RXRZ(4, L); QSA_RXRZ(5, L); QSA_RXRZ(6, L); QSA_RXRZ(7, L);  \
    cnot_gate<7, 6>(ar, ai, lane); cnot_gate<6, 5>(ar, ai, lane);    \
    cnot_gate<5, 4>(ar, ai, lane); cnot_gate<4, 3>(ar, ai, lane);    \
    cnot_gate<3, 2>(ar, ai, lane); cnot_gate<2, 1>(ar, ai, lane);    \
    cnot_gate<1, 0>(ar, ai, lane); } while (0)

  QSA_LAYER(0); QSA_LAYER(1); QSA_LAYER(2); QSA_LAYER(3);

  // PauliZ expectation values: signs[i, idx] = 1 - 2*bit_{7-i}(idx)
  float p[8];
#pragma unroll
  for (int r = 0; r < 8; ++r) p[r] = ar[r] * ar[r] + ai[r] * ai[r];
  const float P = ((p[0] + p[1]) + (p[2] + p[3])) + ((p[4] + p[5]) + (p[6] + p[7]));

  float e[8];
  e[0] = ((lane >> 4) & 1) ? -P : P;
  e[1] = ((lane >> 3) & 1) ? -P : P;
  e[2] = ((lane >> 2) & 1) ? -P : P;
  e[3] = ((lane >> 1) & 1) ? -P : P;
  e[4] = (lane & 1) ? -P : P;
  e[5] = (p[0] + p[1] + p[2] + p[3]) - (p[4] + p[5] + p[6] + p[7]);
  e[6] = (p[0] + p[1] + p[4] + p[5]) - (p[2] + p[3] + p[6] + p[7]);
  e[7] = (p[0] + p[2] + p[4] + p[6]) - (p[1] + p[3] + p[5] + p[7]);

#pragma unroll
  for (int m = 1; m < 32; m <<= 1) {
#pragma unroll
    for (int i = 0; i < 8; ++i) e[i] += sx32(e[i], m);
  }
  // every lane now holds the full ev[0..7] for this sample

  // out[row, :] = ev @ W_out^T + b_out ; lane emits 4 contiguous columns x 8
#pragma unroll
  for (int j = 0; j < 8; ++j) {
    const int c0 = j * 128 + lane * 4;
    float acc[4];
#pragma unroll
    for (int q = 0; q < 4; ++q) {
      const float4 w0 = *(const float4*)(Wout + (size_t)(c0 + q) * 8);
      const float4 w1 = *(const float4*)(Wout + (size_t)(c0 + q) * 8 + 4);
      acc[q] = bout[c0 + q]
             + e[0] * w0.x + e[1] * w0.y + e[2] * w0.z + e[3] * w0.w
             + e[4] * w1.x + e[5] * w1.y + e[6] * w1.z + e[7] * w1.w;
    }
    float4 res = make_float4(acc[0], acc[1], acc[2], acc[3]);
    *(float4*)(out + (size_t)row * DOUT + c0) = res;
  }
}

// ------------------------------- launcher ----------------------------------
extern "C" void kernel_launch(void* const* d_in, const int* in_sizes, int n_in,
                              void* d_out, int out_size, void* d_ws, size_t ws_size,
                              hipStream_t stream) {
  const float* x    = (const float*)d_in[0];  // [B, 1024]
  const float* Win  = (const float*)d_in[1];  // [8, 1024]
  const float* bin  = (const float*)d_in[2];  // [8]
  const float* qw   = (const float*)d_in[3];  // [4, 2, 8]
  const float* Wout = (const float*)d_in[4];  // [1024, 8]
  const float* bout = (const float*)d_in[5];  // [1024]
  float* out = (float*)d_out;                 // [B, 1024]

  const int B = in_sizes[0] / DIN;            // 8192

  float* wpad  = (float*)d_ws;                // 16*1024 floats (64 KB)
  float* xproj = wpad + 16 * DIN;             // B*8 floats (256 KB)

  qsa_pad_win<<<(16 * DIN) / 256, 256, 0, stream>>>(Win, wpad);
  qsa_gemm1_wmma<<<B / 16, 32, 0, stream>>>(x, wpad, bin, xproj);
  qsa_qsim_out<<<B / 8, 256, 0, stream>>>(xproj, qw, Wout, bout, out);
}